// Decoder_63642825392466
// MI455X (gfx1250) — compile-verified
//
#include <hip/hip_runtime.h>
#include <hip/hip_bf16.h>
#include <math.h>

// Problem constants (match reference)
#define Hdim 1024
#define Bdim 64
#define Vdim 32000
#define Tlen 64
#define SOS_TOK 1

#define KB     64      // K-block staged in LDS
#define PITCH  68      // LDS row pitch in floats (68 mod 64 = 4 -> conflict-free b64 reads)
#define NKB    (Hdim / KB)
#define NCOLS  128     // vocab columns per logits block (8 WMMA N-tiles)

typedef float v2f __attribute__((ext_vector_type(2)));
typedef float v8f __attribute__((ext_vector_type(8)));

__device__ __forceinline__ float sigmoidf_(float x) { return 1.0f / (1.0f + expf(-x)); }

// CDNA5 async global->LDS copy (ASYNCcnt-tracked). VDST = LDS byte address,
// VADDR = 64-bit global address (GV mode).
__device__ __forceinline__ void async_b128(uint32_t lds_byte_addr, const float* gaddr) {
    asm volatile("global_load_async_to_lds_b128 %0, %1, off"
                 :: "v"(lds_byte_addr), "v"(gaddr)
                 : "memory");
}
__device__ __forceinline__ void wait_async0() {
    asm volatile("s_wait_asynccnt 0x0" ::: "memory");
}

// -------------------------------------------------------------------------
// init: h0 = encoder_hidden[0], tok = SOS
// -------------------------------------------------------------------------
__global__ void init_kernel(const float* __restrict__ enc_hidden,
                            float* __restrict__ h, int* __restrict__ tok) {
    int i = blockIdx.x * blockDim.x + threadIdx.x;
    if (i < Bdim * Hdim) h[i] = enc_hidden[i];
    if (i < Bdim) tok[i] = SOS_TOK;
}

// -------------------------------------------------------------------------
// GRU step helpers: one "operand set" = A(x), A(h) and the six B tiles.
// Ping-pong two sets so each set is redefined only after consumption
// (no cross-live copies -> no v_mov churn at the loop back-edge).
// -------------------------------------------------------------------------
struct GruOps { v2f ax, ah, b0, b1, b2, b3, b4, b5; };

__device__ __forceinline__ GruOps gru_load(const float* xrow, const float* hrow,
                                           const float* wir, const float* wiz,
                                           const float* win, const float* whr,
                                           const float* whz, const float* whn,
                                           int kk) {
    GruOps o;
    o.ax = *(const v2f*)(xrow + kk);
    o.ah = *(const v2f*)(hrow + kk);
    o.b0 = *(const v2f*)(wir + kk);
    o.b1 = *(const v2f*)(wiz + kk);
    o.b2 = *(const v2f*)(win + kk);
    o.b3 = *(const v2f*)(whr + kk);
    o.b4 = *(const v2f*)(whz + kk);
    o.b5 = *(const v2f*)(whn + kk);
    return o;
}

__device__ __forceinline__ void gru_wmma(const GruOps& o,
                                         v8f& acc_ir, v8f& acc_iz, v8f& acc_in,
                                         v8f& acc_hr, v8f& acc_hz, v8f& acc_hn) {
    acc_ir = __builtin_amdgcn_wmma_f32_16x16x4_f32(false, o.ax, false, o.b0, (short)0, acc_ir, false, false);
    acc_iz = __builtin_amdgcn_wmma_f32_16x16x4_f32(false, o.ax, false, o.b1, (short)0, acc_iz, false, false);
    acc_in = __builtin_amdgcn_wmma_f32_16x16x4_f32(false, o.ax, false, o.b2, (short)0, acc_in, false, false);
    acc_hr = __builtin_amdgcn_wmma_f32_16x16x4_f32(false, o.ah, false, o.b3, (short)0, acc_hr, false, false);
    acc_hz = __builtin_amdgcn_wmma_f32_16x16x4_f32(false, o.ah, false, o.b4, (short)0, acc_hz, false, false);
    acc_hn = __builtin_amdgcn_wmma_f32_16x16x4_f32(false, o.ah, false, o.b5, (short)0, acc_hn, false, false);
}

// -------------------------------------------------------------------------
// GRU step: one wave per 16x16 tile of h_new.
// A (16x4 f32): lanes 0-15: M=lane, K=k0+{0,1}; lanes 16-31: K=k0+{2,3}
// B (4x16 f32): mirrored; C (16x16): VGPR g -> M=g+8*(lane>>4), N=lane&15
// waves_per_eu(1): single-wave workgroup; give RA the full register file.
// -------------------------------------------------------------------------
__global__ void
__attribute__((amdgpu_flat_work_group_size(32, 32), amdgpu_waves_per_eu(1)))
gru_step_kernel(const float* __restrict__ emb,
                const int*  __restrict__ tok,
                const float* __restrict__ hprev,
                const float* __restrict__ Wih,
                const float* __restrict__ Whh,
                const float* __restrict__ bih,
                const float* __restrict__ bhh,
                float* __restrict__ hnext) {
    const int lane = threadIdx.x;     // 0..31, full wave, EXEC all ones
    const int r    = lane & 15;
    const int half = lane >> 4;
    const int n0   = blockIdx.x * 16; // hidden-column tile
    const int m0   = blockIdx.y * 16; // batch-row tile

    const int tk = tok[m0 + r];
    const float* xrow = emb   + (size_t)tk * Hdim;          // embedding gather
    const float* hrow = hprev + (size_t)(m0 + r) * Hdim;

    // B tiles: B[k][n] = W[n0+n][k]  (gi = x @ W^T)
    const float* wir = Wih + (size_t)(0 * Hdim + n0 + r) * Hdim;
    const float* wiz = Wih + (size_t)(1 * Hdim + n0 + r) * Hdim;
    const float* win = Wih + (size_t)(2 * Hdim + n0 + r) * Hdim;
    const float* whr = Whh + (size_t)(0 * Hdim + n0 + r) * Hdim;
    const float* whz = Whh + (size_t)(1 * Hdim + n0 + r) * Hdim;
    const float* whn = Whh + (size_t)(2 * Hdim + n0 + r) * Hdim;

    v8f acc_ir = {}, acc_iz = {}, acc_in = {}, acc_hr = {}, acc_hz = {}, acc_hn = {};

    const int kb = 2 * half;
    GruOps s0 = gru_load(xrow, hrow, wir, wiz, win, whr, whz, whn, kb);

    // ping-pong unroll-by-2: 127 iterations x 2 chunks, + 2 epilogue chunks
    for (int k0 = 0; k0 < Hdim - 8; k0 += 8) {
        GruOps s1 = gru_load(xrow, hrow, wir, wiz, win, whr, whz, whn, k0 + 4 + kb);
        gru_wmma(s0, acc_ir, acc_iz, acc_in, acc_hr, acc_hz, acc_hn);
        s0 = gru_load(xrow, hrow, wir, wiz, win, whr, whz, whn, k0 + 8 + kb);
        gru_wmma(s1, acc_ir, acc_iz, acc_in, acc_hr, acc_hz, acc_hn);
    }
    {
        GruOps s1 = gru_load(xrow, hrow, wir, wiz, win, whr, whz, whn, Hdim - 4 + kb);
        gru_wmma(s0, acc_ir, acc_iz, acc_in, acc_hr, acc_hz, acc_hn);
        gru_wmma(s1, acc_ir, acc_iz, acc_in, acc_hr, acc_hz, acc_hn);
    }

    const int N  = n0 + r;
    const float bir = bih[N],          bhr_ = bhh[N];
    const float biz = bih[Hdim + N],   bhz_ = bhh[Hdim + N];
    const float bin = bih[2*Hdim + N], bhn_ = bhh[2*Hdim + N];

#pragma unroll
    for (int g = 0; g < 8; ++g) {
        const int M = m0 + g + 8 * half;   // batch row
        float ir = acc_ir[g] + bir, hr = acc_hr[g] + bhr_;
        float iz = acc_iz[g] + biz, hz = acc_hz[g] + bhz_;
        float in_ = acc_in[g] + bin, hn = acc_hn[g] + bhn_;
        float rg = sigmoidf_(ir + hr);
        float zg = sigmoidf_(iz + hz);
        float ng = tanhf(in_ + rg * hn);
        float hp = hprev[(size_t)M * Hdim + N];
        hnext[(size_t)M * Hdim + N] = (1.0f - zg) * ng + zg * hp;
    }
}

// -------------------------------------------------------------------------
// logits: 128-thread block (4 waves) computes a 64(batch) x 128(vocab) tile.
// W_out and h K-blocks staged in LDS via async global->LDS b128 copies,
// double-buffered; WMMA operands come from conflict-free ds_load_b64.
// LDS: 2*128 rows (B) + 2*64 rows (A) at pitch 68 = ~104 KB (<320 KB WGP LDS).
// -------------------------------------------------------------------------
__global__ void
__attribute__((amdgpu_flat_work_group_size(128, 128), amdgpu_waves_per_eu(1)))
logits_kernel(const float* __restrict__ h,
              const float* __restrict__ Wout,
              const float* __restrict__ bout,
              float* __restrict__ out, int t) {
    __shared__ float Ash[2][64 * PITCH];    // h rows 0..63, K-block
    __shared__ float Bsh[2][NCOLS * PITCH]; // W_out rows v0..v0+127, K-block

    const int tid  = threadIdx.x;
    const int lane = tid & 31;
    const int wid  = tid >> 5;            // wave id -> batch m-tile
    const int r    = lane & 15;
    const int half = lane >> 4;
    const int v0   = blockIdx.x * NCOLS;  // vocab macro-tile base
    const int m0   = wid * 16;

    // cooperative-copy geometry: 16 threads per row segment, 8 rows per pass
    const int rsub = tid >> 4;            // 0..7
    const int c4   = (tid & 15) * 4;      // float col within K-block

    // ---- issue async copies for K-block 0 into buffer 0 ----
#pragma unroll
    for (int pass = 0; pass < NCOLS / 8; ++pass) {
        const int row = pass * 8 + rsub;
        async_b128((uint32_t)(uintptr_t)(&Bsh[0][row * PITCH + c4]),
                   Wout + (size_t)(v0 + row) * Hdim + c4);
    }
#pragma unroll
    for (int pass = 0; pass < 8; ++pass) {
        const int row = pass * 8 + rsub;
        async_b128((uint32_t)(uintptr_t)(&Ash[0][row * PITCH + c4]),
                   h + (size_t)row * Hdim + c4);
    }
    wait_async0();
    __syncthreads();

    v8f acc[8] = {{}, {}, {}, {}, {}, {}, {}, {}};
    int p = 0;

    for (int kb = 0; kb < NKB; ++kb) {
        // ---- prefetch next K-block into the other buffer ----
        if (kb + 1 < NKB) {
            const int kbase = (kb + 1) * KB;
#pragma unroll
            for (int pass = 0; pass < NCOLS / 8; ++pass) {
                const int row = pass * 8 + rsub;
                async_b128((uint32_t)(uintptr_t)(&Bsh[p ^ 1][row * PITCH + c4]),
                           Wout + (size_t)(v0 + row) * Hdim + kbase + c4);
            }
#pragma unroll
            for (int pass = 0; pass < 8; ++pass) {
                const int row = pass * 8 + rsub;
                async_b128((uint32_t)(uintptr_t)(&Ash[p ^ 1][row * PITCH + c4]),
                           h + (size_t)row * Hdim + kbase + c4);
            }
        }

        // ---- compute on current buffer: 16 k-steps x 8 WMMA ----
        const float* Ap = &Ash[p][(m0 + r) * PITCH];
        const float* Bp = &Bsh[p][r * PITCH];
#pragma unroll
        for (int i = 0; i < KB / 4; ++i) {
            const int kk = i * 4 + 2 * half;
            v2f a = *(const v2f*)(Ap + kk);
#pragma unroll
            for (int j = 0; j < 8; ++j) {
                v2f b = *(const v2f*)(Bp + (j * 16) * PITCH + kk);
                acc[j] = __builtin_amdgcn_wmma_f32_16x16x4_f32(
                    false, a, false, b, (short)0, acc[j], false, false);
            }
        }

        wait_async0();     // prefetch for next block has landed (per-wave)
        __syncthreads();   // all waves done reading buffer p, data visible
        p ^= 1;
    }

    // ---- epilogue: add bias, store raw logits into d_out[b, t, :] ----
#pragma unroll
    for (int j = 0; j < 8; ++j) {
        const int col = v0 + j * 16 + r;
        const float bb = bout[col];
#pragma unroll
        for (int g = 0; g < 8; ++g) {
            const int M = m0 + g + 8 * half;
            out[(size_t)M * Tlen * Vdim + (size_t)t * Vdim + col] = acc[j][g] + bb;
        }
    }
}

// -------------------------------------------------------------------------
// per-row: argmax (first-index tie-break, like jnp.argmax) -> tok,
// then log_softmax rewrite of the row in place.
// -------------------------------------------------------------------------
__global__ void softmax_argmax_kernel(float* __restrict__ out,
                                      int* __restrict__ tok, int t) {
    __shared__ float sval[256];
    __shared__ int   sidx[256];
    const int b   = blockIdx.x;
    const int tid = threadIdx.x;
    float* row = out + (size_t)b * Tlen * Vdim + (size_t)t * Vdim;

    // pass 1: max + argmax
    float m = -INFINITY;
    int   mi = 0x7fffffff;
    for (int i = tid; i < Vdim; i += 256) {
        float v = row[i];
        if (v > m || (v == m && i < mi)) { m = v; mi = i; }
    }
    sval[tid] = m; sidx[tid] = mi;
    __syncthreads();
    for (int s = 128; s > 0; s >>= 1) {
        if (tid < s) {
            float v2 = sval[tid + s]; int i2 = sidx[tid + s];
            if (v2 > sval[tid] || (v2 == sval[tid] && i2 < sidx[tid])) {
                sval[tid] = v2; sidx[tid] = i2;
            }
        }
        __syncthreads();
    }
    const float rmax = sval[0];
    const int   amax = sidx[0];
    __syncthreads();

    // pass 2: logsumexp
    float s = 0.0f;
    for (int i = tid; i < Vdim; i += 256) s += expf(row[i] - rmax);
    sval[tid] = s;
    __syncthreads();
    for (int st = 128; st > 0; st >>= 1) {
        if (tid < st) sval[tid] += sval[tid + st];
        __syncthreads();
    }
    const float lse = rmax + logf(sval[0]);
    if (tid == 0) tok[b] = amax;

    // pass 3: rewrite as log-probs
    for (int i = tid; i < Vdim; i += 256) row[i] = row[i] - lse;
}

// -------------------------------------------------------------------------
extern "C" void kernel_launch(void* const* d_in, const int* in_sizes, int n_in,
                              void* d_out, int out_size, void* d_ws, size_t ws_size,
                              hipStream_t stream) {
    (void)in_sizes; (void)n_in; (void)out_size; (void)ws_size;
    // inputs: 0 encoder_outputs (unused, no attention in reference),
    // 1 encoder_hidden, 2 emb, 3 W_ih, 4 W_hh, 5 b_ih, 6 b_hh, 7 W_out, 8 b_out
    const float* enc_hidden = (const float*)d_in[1];
    const float* emb        = (const float*)d_in[2];
    const float* Wih        = (const float*)d_in[3];
    const float* Whh        = (const float*)d_in[4];
    const float* bih        = (const float*)d_in[5];
    const float* bhh        = (const float*)d_in[6];
    const float* Wout       = (const float*)d_in[7];
    const float* bout       = (const float*)d_in[8];
    float* out = (float*)d_out;

    // workspace: two hidden-state buffers (ping-pong) + token feedback array
    float* hA  = (float*)d_ws;
    float* hB  = hA + Bdim * Hdim;
    int*   tok = (int*)(hB + Bdim * Hdim);

    init_kernel<<<(Bdim * Hdim + 255) / 256, 256, 0, stream>>>(enc_hidden, hA, tok);

    float* cur = hA;
    float* nxt = hB;
    for (int t = 0; t < Tlen; ++t) {
        gru_step_kernel<<<dim3(Hdim / 16, Bdim / 16), 32, 0, stream>>>(
            emb, tok, cur, Wih, Whh, bih, bhh, nxt);
        logits_kernel<<<Vdim / NCOLS, 128, 0, stream>>>(nxt, Wout, bout, out, t);
        softmax_argmax_kernel<<<Bdim, 256, 0, stream>>>(out, tok, t);
        float* tmp = cur; cur = nxt; nxt = tmp;
    }

    // append final hidden state [1, B, H] after the [B, T, V] log-probs
    (void)hipMemcpyAsync(out + (size_t)Bdim * Tlen * Vdim, cur,
                         (size_t)Bdim * Hdim * sizeof(float),
                         hipMemcpyDeviceToDevice, stream);
}